// DiscriminativeLoss_53584011985667
// MI455X (gfx1250) — compile-verified
//
#include <hip/hip_runtime.h>
#include <math.h>
#include <stdint.h>

#define BS 8
#define NF 32
#define MAXC 4
#define NLOC 65536
#define K_ACTIVE 2
#define DELTA_VAR 0.5f
#define DELTA_DIST 1.5f
#define ALPHA 1.0f
#define BETA 1.0f
#define GAMMA 1.0e-4f

#define NBPB 64              // blocks per batch (pass1 & pass2)
#define SPAN (NLOC / NBPB)   // 1024 locations per block
#define TN 128               // locations per LDS tile (pass1)
#define XPITCH 132           // LDS row pitch (floats): 16B aligned, no bank conflicts
#define TPITCH 132

typedef float v2f __attribute__((ext_vector_type(2)));
typedef float v8f __attribute__((ext_vector_type(8)));

// workspace layout (float offsets)
#define WS_COUNTS 0      // [BS][MAXC]
#define WS_SUMS   32     // [BS][NF][MAXC]
#define WS_MEANS  1056   // [BS][NF][MAXC]
#define WS_MSQ    2080   // [BS][MAXC]  ||mean||^2
#define WS_CVAR   2112   // [BS][MAXC]  sum of hinge^2 over members
#define WS_MISC   2144   // beta*l_dist + gamma*l_reg
#define WS_TOTAL  2176

// Direct global->LDS async copy (CDNA5 TDM-adjacent path, ASYNCcnt-tracked).
// lds_off = 32-bit LDS byte offset (low 32 bits of flat address of a __shared__ object).
__device__ __forceinline__ void async_g2l_b128(unsigned lds_off, const void* gaddr) {
    asm volatile("global_load_async_to_lds_b128 %0, %1, off"
                 :: "v"(lds_off), "v"(gaddr)
                 : "memory");
}
__device__ __forceinline__ void wait_asynccnt0() {
    asm volatile("s_wait_asynccnt 0x0" ::: "memory");
}

__global__ void k_zero(float* __restrict__ ws) {
    for (int i = threadIdx.x; i < WS_TOTAL; i += 256) ws[i] = 0.0f;
}

// ---------------- Pass 1: sums[b,f,c] and counts[b,c] via V_WMMA_F32_16X16X4_F32
__global__ __launch_bounds__(256)
void k_pass1(const float* __restrict__ inp, const float* __restrict__ tgt,
             float* __restrict__ ws) {
    __shared__ float xs[NF * XPITCH];      // input tile  32 x 128 (pitch 132)
    __shared__ float ts[16 * TPITCH];      // target tile 16 x 128 (rows 4..15 stay zero)
    __shared__ float lsums[NF * MAXC];
    __shared__ float lcounts[MAXC];

    const int tid = threadIdx.x;
    const int b   = blockIdx.x / NBPB;
    const int blk = blockIdx.x % NBPB;
    const int nstart = blk * SPAN;

    for (int i = tid; i < 16 * TPITCH; i += 256) ts[i] = 0.0f;
    for (int i = tid; i < NF * MAXC; i += 256) lsums[i] = 0.0f;
    if (tid < MAXC) lcounts[tid] = 0.0f;
    __syncthreads();

    const int lane = tid & 31;
    const int wave = tid >> 5;       // 0..7, each wave owns 16 locations of the tile
    const int half = lane >> 4;      // WMMA K-half select
    const int mrow = lane & 15;      // M index (A) / N index (B)

    const int xr = tid >> 3;         // staging: input row 0..31
    const int xq = tid & 7;          // 8 threads per row
    const int trow = tid >> 5;       // staging: target row (tid<128 -> 0..3)
    const int tcol = (tid & 31) * 4;

    v8f acc0 = {};                   // f = 0..15
    v8f acc1 = {};                   // f = 16..31
    float cacc = 0.0f;

    const float* xbase = inp + (size_t)b * NF * NLOC;
    const float* tbase = tgt + (size_t)b * MAXC * NLOC;

    // precompute the 4 LDS byte offsets this thread stages into (constant per thread)
    unsigned xs_off[4];
    #pragma unroll
    for (int k = 0; k < 4; ++k) {
        const int col = (xq + k * 8) * 4;
        xs_off[k] = (unsigned)(uintptr_t)(&xs[xr * XPITCH + col]);
    }

    for (int t0 = 0; t0 < SPAN; t0 += TN) {
        const int n0 = nstart + t0;
        // stage input tile: direct global->LDS async copy (no VGPR round-trip)
        #pragma unroll
        for (int k = 0; k < 4; ++k) {
            const int col = (xq + k * 8) * 4;
            async_g2l_b128(xs_off[k], xbase + (size_t)xr * NLOC + n0 + col);
        }
        // stage target tile rows 0..3 through registers, fold in counts
        if (tid < 128) {
            const float4 v = *(const float4*)(tbase + (size_t)trow * NLOC + n0 + tcol);
            *(float4*)(&ts[trow * TPITCH + tcol]) = v;
            cacc += v.x + v.y + v.z + v.w;
        }
        // prefetch next tile into cache hierarchy
        if (t0 + TN < SPAN)
            __builtin_prefetch(xbase + (size_t)xr * NLOC + n0 + TN + xq * 16, 0, 0);
        wait_asynccnt0();
        __syncthreads();

        // WMMA: A = input (16xK=4 locations), B = target^T (4 x 16 clusters, cols>=4 zero)
        #pragma unroll
        for (int s = 0; s < 4; ++s) {
            const int lb = wave * 16 + s * 4 + 2 * half;   // even -> 8B aligned ds_read_b64
            const v2f a0 = *(const v2f*)(&xs[mrow * XPITCH + lb]);
            const v2f a1 = *(const v2f*)(&xs[(16 + mrow) * XPITCH + lb]);
            const v2f bf = *(const v2f*)(&ts[mrow * TPITCH + lb]);
            acc0 = __builtin_amdgcn_wmma_f32_16x16x4_f32(
                       false, a0, false, bf, (short)0, acc0, false, false);
            acc1 = __builtin_amdgcn_wmma_f32_16x16x4_f32(
                       false, a1, false, bf, (short)0, acc1, false, false);
        }
        __syncthreads();
    }

    // C layout: lane (mrow,half), reg r -> M = r + 8*half, N = mrow
    if (mrow < MAXC) {
        #pragma unroll
        for (int r = 0; r < 8; ++r) {
            const int f = r + 8 * half;
            atomicAdd(&lsums[f * MAXC + mrow], acc0[r]);
            atomicAdd(&lsums[(16 + f) * MAXC + mrow], acc1[r]);
        }
    }
    if (tid < 128) atomicAdd(&lcounts[trow], cacc);
    __syncthreads();

    if (tid < NF * MAXC) atomicAdd(&ws[WS_SUMS + b * NF * MAXC + tid], lsums[tid]);
    if (tid < MAXC)      atomicAdd(&ws[WS_COUNTS + b * MAXC + tid], lcounts[tid]);
}

// ---------------- means, ||m||^2, l_dist, l_reg (tiny)
__global__ void k_means(const int* __restrict__ ncl, float* __restrict__ ws) {
    const int tid = threadIdx.x;
    for (int i = tid; i < BS * NF * MAXC; i += 256) {
        const int b = i / (NF * MAXC);
        const int c = i & (MAXC - 1);
        const float cnt = ws[WS_COUNTS + b * MAXC + c];
        ws[WS_MEANS + i] = (cnt > 0.0f) ? ws[WS_SUMS + i] / cnt : 0.0f;
    }
    __syncthreads();
    for (int i = tid; i < BS * MAXC; i += 256) {
        const int b = i >> 2, c = i & 3;
        float s = 0.0f;
        for (int f = 0; f < NF; ++f) {
            const float m = ws[WS_MEANS + (b * NF + f) * MAXC + c];
            s = fmaf(m, m, s);
        }
        ws[WS_MSQ + i] = s;
    }
    __syncthreads();
    if (tid == 0) {
        float ldist = 0.0f, lreg = 0.0f;
        for (int b = 0; b < BS; ++b) {
            const float nc = (float)ncl[b];
            float cd = 0.0f;
            for (int ca = 0; ca < K_ACTIVE; ++ca)
                for (int cb = 0; cb < K_ACTIVE; ++cb) {
                    if (ca == cb) continue;
                    float dsq = 0.0f;
                    for (int f = 0; f < NF; ++f) {
                        const float d = ws[WS_MEANS + (b * NF + f) * MAXC + ca] -
                                        ws[WS_MEANS + (b * NF + f) * MAXC + cb];
                        dsq = fmaf(d, d, dsq);
                    }
                    const float dn = (dsq > 0.0f) ? sqrtf(dsq) : 0.0f;
                    float h = 2.0f * DELTA_DIST - dn;
                    h = (h > 0.0f) ? h : 0.0f;
                    cd += h * h;
                }
            ldist += cd / (2.0f * nc * (nc - 1.0f));
            float rg = 0.0f;
            for (int c = 0; c < K_ACTIVE; ++c) {
                const float m2 = ws[WS_MSQ + b * MAXC + c];
                rg += (m2 > 0.0f) ? sqrtf(m2) : 0.0f;
            }
            lreg += rg / (float)K_ACTIVE;
        }
        ws[WS_MISC] = BETA * (ldist / (float)BS) + GAMMA * (lreg / (float)BS);
    }
}

// ---------------- Pass 2: per-cluster variance sums (coalesced, L2-resident reuse)
__global__ __launch_bounds__(256)
void k_pass2(const float* __restrict__ inp, const float* __restrict__ tgt,
             float* __restrict__ ws) {
    __shared__ float ms[NF * MAXC];
    __shared__ float msq_s[MAXC];
    __shared__ float cv[MAXC];

    const int tid = threadIdx.x;
    const int b   = blockIdx.x / NBPB;
    const int blk = blockIdx.x % NBPB;
    const int n0  = blk * SPAN + tid;   // thread handles n0 + 256*j, j=0..3

    if (tid < NF * MAXC) ms[tid] = ws[WS_MEANS + b * NF * MAXC + tid];
    if (tid < MAXC) { msq_s[tid] = ws[WS_MSQ + b * MAXC + tid]; cv[tid] = 0.0f; }
    __syncthreads();

    float sq[4] = {0.0f, 0.0f, 0.0f, 0.0f};
    float dot[4][MAXC] = {};
    const float* xb = inp + (size_t)b * NF * NLOC + n0;

    #pragma unroll 4
    for (int f = 0; f < NF; ++f) {
        const float m0 = ms[f * MAXC + 0], m1 = ms[f * MAXC + 1];
        const float m2 = ms[f * MAXC + 2], m3 = ms[f * MAXC + 3];
        #pragma unroll
        for (int j = 0; j < 4; ++j) {
            const float v = xb[(size_t)f * NLOC + j * 256];
            sq[j]     = fmaf(v, v,  sq[j]);
            dot[j][0] = fmaf(v, m0, dot[j][0]);
            dot[j][1] = fmaf(v, m1, dot[j][1]);
            dot[j][2] = fmaf(v, m2, dot[j][2]);
            dot[j][3] = fmaf(v, m3, dot[j][3]);
        }
    }

    float vacc[MAXC] = {0.0f, 0.0f, 0.0f, 0.0f};
    const float* tb = tgt + (size_t)b * MAXC * NLOC + n0;
    #pragma unroll
    for (int j = 0; j < 4; ++j) {
        #pragma unroll
        for (int c = 0; c < MAXC; ++c) {
            const float t = tb[(size_t)c * NLOC + j * 256];
            float d2 = sq[j] - 2.0f * dot[j][c] + msq_s[c];
            d2 = (d2 > 0.0f) ? d2 : 0.0f;
            const float dist = sqrtf(d2);
            float h = dist - DELTA_VAR;
            h = (h > 0.0f) ? h : 0.0f;
            vacc[c] = fmaf(h * h, t, vacc[c]);
        }
    }
    #pragma unroll
    for (int c = 0; c < MAXC; ++c) atomicAdd(&cv[c], vacc[c]);
    __syncthreads();
    if (tid < MAXC) atomicAdd(&ws[WS_CVAR + b * MAXC + tid], cv[tid]);
}

// ---------------- final scalar
__global__ void k_final(const int* __restrict__ ncl, const float* __restrict__ ws,
                        float* __restrict__ out) {
    if (threadIdx.x == 0 && blockIdx.x == 0) {
        float lvar = 0.0f;
        for (int b = 0; b < BS; ++b) {
            const float nc = (float)ncl[b];
            float s = 0.0f;
            for (int c = 0; c < MAXC; ++c) {
                const float cnt = ws[WS_COUNTS + b * MAXC + c];
                if (cnt > 0.0f) s += ws[WS_CVAR + b * MAXC + c] / cnt;
            }
            lvar += s / nc;
        }
        lvar /= (float)BS;
        out[0] = ALPHA * lvar + ws[WS_MISC];
    }
}

extern "C" void kernel_launch(void* const* d_in, const int* in_sizes, int n_in,
                              void* d_out, int out_size, void* d_ws, size_t ws_size,
                              hipStream_t stream) {
    (void)in_sizes; (void)n_in; (void)out_size; (void)ws_size;
    const float* inp = (const float*)d_in[0];
    const float* tgt = (const float*)d_in[1];
    const int*   ncl = (const int*)d_in[2];
    float* out = (float*)d_out;
    float* ws  = (float*)d_ws;

    hipLaunchKernelGGL(k_zero,  dim3(1),         dim3(256), 0, stream, ws);
    hipLaunchKernelGGL(k_pass1, dim3(BS * NBPB), dim3(256), 0, stream, inp, tgt, ws);
    hipLaunchKernelGGL(k_means, dim3(1),         dim3(256), 0, stream, ncl, ws);
    hipLaunchKernelGGL(k_pass2, dim3(BS * NBPB), dim3(256), 0, stream, inp, tgt, ws);
    hipLaunchKernelGGL(k_final, dim3(1),         dim3(32),  0, stream, ncl, ws, out);
}